// ContinuousGaborFilterBanks_36180804501585
// MI455X (gfx1250) — compile-verified
//
#include <hip/hip_runtime.h>
#include <math.h>

#define RES     256
#define GRID_N  (RES * RES)       // 65536
#define NPTS    2048
#define KNN     30
#define NFREQ   9
#define AATTR   4
#define NCH     (1 + NFREQ * AATTR)   // 37
#define TWO_PI_F 6.28318530717958647692f

typedef __attribute__((ext_vector_type(2))) float v2f;
typedef __attribute__((ext_vector_type(8))) float v8f;

// ---------------------------------------------------------------------------
// Kernel 1: per-column max of rad[2048][4], then rad_n = rad / max (into ws).
// ---------------------------------------------------------------------------
__global__ __launch_bounds__(256) void rad_normalize_kernel(
    const float* __restrict__ rad, float* __restrict__ rad_n) {
  __shared__ float sm[256 * 4];
  const int tid = threadIdx.x;
  float m[4] = {0.f, 0.f, 0.f, 0.f};  // rad > 0 always
  for (int r = tid; r < NPTS; r += 256) {
#pragma unroll
    for (int c = 0; c < 4; ++c) m[c] = fmaxf(m[c], rad[r * 4 + c]);
  }
#pragma unroll
  for (int c = 0; c < 4; ++c) sm[tid * 4 + c] = m[c];
  __syncthreads();
  for (int s = 128; s > 0; s >>= 1) {
    if (tid < s) {
#pragma unroll
      for (int c = 0; c < 4; ++c)
        sm[tid * 4 + c] = fmaxf(sm[tid * 4 + c], sm[(tid + s) * 4 + c]);
    }
    __syncthreads();
  }
  float inv[4];
#pragma unroll
  for (int c = 0; c < 4; ++c) inv[c] = 1.0f / sm[c];
  for (int r = tid; r < NPTS; r += 256) {
#pragma unroll
    for (int c = 0; c < 4; ++c) rad_n[r * 4 + c] = rad[r * 4 + c] * inv[c];
  }
}

// ---------------------------------------------------------------------------
// Kernel 2: fused WMMA-distance kNN + Gabor filter-bank evaluation.
// Block = 256 threads (8 waves). Each wave owns 16 grid points; block owns 128.
// ---------------------------------------------------------------------------
__global__ __launch_bounds__(256) void gabor_knn_kernel(
    const float* __restrict__ pts, const float* __restrict__ rad_n,
    const float* __restrict__ ksig, float* __restrict__ out) {
  __shared__ float    s_pts[NPTS * 2];     // 16 KB  candidate points
  __shared__ float    s_tile[8][16 * 16];  //  8 KB  per-wave D tiles
  __shared__ unsigned s_top[128][KNN];     // 15 KB  per-grid-point top-30 keys

  const int tid   = threadIdx.x;
  const int lane  = tid & 31;
  const int wave  = tid >> 5;
  const int gBase = blockIdx.x * 128;

  __builtin_prefetch(rad_n, 0, 1);  // global_prefetch_b8 toward L2

  // Stage all candidate points + init top lists.
  for (int i = tid; i < NPTS * 2; i += 256) s_pts[i] = pts[i];
  {
    unsigned* flat = &s_top[0][0];
    for (int i = tid; i < 128 * KNN; i += 256) flat[i] = 0xFFFFFFFFu;
  }
  __syncthreads();

  // ---- WMMA A operand: 16x4 f32, lanes 0-15 hold K=0,1; lanes 16-31 K=2,3.
  // A[m,0..3] = {-2*gx, -2*gy, 1, 0}  so  D = -2 g.p + |p|^2  with
  // B[0..3,n] = {px, py, |p|^2, 0}.
  const int   m  = lane & 15;
  const int   hi = lane >> 4;
  const int   gA = gBase + wave * 16 + m;
  const float gx = (float)((gA & 255) + 1) * (1.0f / 257.0f);
  const float gy = (float)((gA >> 8) + 1) * (1.0f / 257.0f);
  v2f A;
  A.x = hi ? 1.0f : -2.0f * gx;  // K=0 (or K=2 ones-column)
  A.y = hi ? 0.0f : -2.0f * gy;  // K=1 (or K=3 zero)
  const float gsq = gx * gx + gy * gy;

  float*    tile  = s_tile[wave];
  unsigned* mytop = s_top[wave * 16 + m];  // used only by lanes 0-15

  unsigned maxkey = 0xFFFFFFFFu;
  int      maxpos = 0;

  for (int c0 = 0; c0 < NPTS; c0 += 16) {
    const int   n  = c0 + m;
    const float px = s_pts[n * 2 + 0];
    const float py = s_pts[n * 2 + 1];
    v2f B;
    B.x = hi ? (px * px + py * py) : px;  // K=0 row / K=2 row (|p|^2)
    B.y = hi ? 0.0f : py;                 // K=1 row / K=3 row (zero)
    v8f C = {0.f, 0.f, 0.f, 0.f, 0.f, 0.f, 0.f, 0.f};
    C = __builtin_amdgcn_wmma_f32_16x16x4_f32(false, A, false, B, (short)0, C,
                                              false, false);
    // D layout: VGPR v, lanes 0-15 -> M=v, lanes 16-31 -> M=v+8; N = lane%16.
#pragma unroll
    for (int v = 0; v < 8; ++v) tile[(v + 8 * hi) * 16 + m] = C[v];

    // Lanes 0-15: scan the 16 candidates of row m, keep top-30 keys.
    if (hi == 0) {
      for (int t = 0; t < 16; ++t) {
        float d2 = fmaxf(tile[m * 16 + t] + gsq, 0.0f);
        unsigned key =
            (__float_as_uint(d2) & 0xFFFFF800u) | (unsigned)(c0 + t);
        if (key < maxkey) {
          mytop[maxpos] = key;
          unsigned mk = 0u;
          int      mp = 0;
          for (int j = 0; j < KNN; ++j) {
            unsigned kj = mytop[j];
            if (kj > mk) { mk = kj; mp = j; }
          }
          maxkey = mk;
          maxpos = mp;
        }
      }
    }
  }
  __syncthreads();

  // ---- Phase 3: Gabor evaluation over the selected 30 neighbors.
  if (tid < 128) {
    const int   g   = gBase + tid;
    const float qx  = (float)((g & 255) + 1) * (1.0f / 257.0f);
    const float qy  = (float)((g >> 8) + 1) * (1.0f / 257.0f);
    const float sig = ksig[0] * 0.5f;  // kernel_sigma / UPSCALE
    const float inv2s2 = 0.5f / (sig * sig);

    float accS = 0.f;
    float acc[NFREQ * AATTR];
#pragma unroll
    for (int c = 0; c < NFREQ * AATTR; ++c) acc[c] = 0.f;

    const unsigned* top = s_top[tid];
    const float F[3] = {-3.f, 0.f, 3.f};

    for (int k = 0; k < KNN; ++k) {
      const unsigned idx = top[k] & 0x7FFu;
      const float dx = s_pts[idx * 2 + 0] - qx;
      const float dy = s_pts[idx * 2 + 1] - qy;
      const float sp = __expf(-(dx * dx + dy * dy) * inv2s2);
      accS += sp;
      float ra[AATTR];
#pragma unroll
      for (int a = 0; a < AATTR; ++a) ra[a] = rad_n[idx * 4 + a];
      float pj[NFREQ];
#pragma unroll
      for (int s = 0; s < 3; ++s)
#pragma unroll
        for (int r = 0; r < 3; ++r)
          pj[s * 3 + r] = __cosf(TWO_PI_F * (dx * F[r] + dy * F[s])) * sp;
      // rv[c] = power[c % 9] * rad[c % 4]   (tile periodicity mismatch)
#pragma unroll
      for (int c = 0; c < NFREQ * AATTR; ++c)
        acc[c] += pj[c % NFREQ] * ra[c % AATTR];
    }
    const float invK = 1.0f / (float)KNN;
    out[g] = accS * invK;
#pragma unroll
    for (int c = 0; c < NFREQ * AATTR; ++c)
      out[(1 + c) * GRID_N + g] = acc[c] * invK;
  }
}

// ---------------------------------------------------------------------------
extern "C" void kernel_launch(void* const* d_in, const int* in_sizes, int n_in,
                              void* d_out, int out_size, void* d_ws,
                              size_t ws_size, hipStream_t stream) {
  const float* pts  = (const float*)d_in[0];  // [2048, 2]
  const float* rad  = (const float*)d_in[1];  // [2048, 4]
  const float* ksig = (const float*)d_in[2];  // [1]
  float*       out  = (float*)d_out;          // [37, 256, 256]
  float*       radn = (float*)d_ws;           // [2048, 4] scratch

  rad_normalize_kernel<<<1, 256, 0, stream>>>(rad, radn);
  gabor_knn_kernel<<<GRID_N / 128, 256, 0, stream>>>(pts, radn, ksig, out);
}